// SubspaceLinopFactory_81990925680757
// MI455X (gfx1250) — compile-verified
//
#include <hip/hip_runtime.h>

// ---------------------------------------------------------------------------
// Subspace linop normal operator for MI455X (gfx1250, wave32).
//   pipeline: expand+pad -> FFT rows -> FFT cols -> 5x5 subspace mix (in place)
//             -> IFFT cols -> IFFT rows -> crop + conj-coil-combine.
//   FFT512 = four-step 16x32 decomposition; both sub-DFTs run on the matrix
//   units via V_WMMA_F32_16X16X4_F32 (complex GEMM = 4 real WMMA chains; the
//   -Ai*Bi term uses a pre-negated imaginary fragment since float WMMA has no
//   A/B NEG modifier).
//   Whole intermediate state (126 MB) is L2-resident on MI455X (192 MB L2).
// ---------------------------------------------------------------------------

typedef float v2f __attribute__((ext_vector_type(2)));
typedef float v8f __attribute__((ext_vector_type(8)));

#define A_SUB   5
#define C_COIL  12
#define H_IMG   256
#define W_IMG   256
#define GH      512
#define GW      512
#define NPLANE  (C_COIL * A_SUB)       // 60 complex images
#define PLANE_ELEMS (GH * GW)          // 262144
#define PI_F    3.14159265358979323846f

// ---------------------------------------------------------------------------
// K1: coil expansion + centered zero-pad.  planes indexed p = c*A_SUB + a.
// ---------------------------------------------------------------------------
__global__ void expand_pad_kernel(const float* __restrict__ xr, const float* __restrict__ xi,
                                  const float* __restrict__ mr, const float* __restrict__ mi,
                                  float* __restrict__ wsR, float* __restrict__ wsI) {
  int tid = blockIdx.x * blockDim.x + threadIdx.x;      // over NPLANE*PLANE_ELEMS
  int pix = tid & (PLANE_ELEMS - 1);
  int p   = tid >> 18;                                  // /262144
  int c = p / A_SUB, a = p - c * A_SUB;
  int gh = pix >> 9, gw = pix & 511;
  int h = gh - 128, w = gw - 128;                       // centered pad offset
  float outr = 0.0f, outi = 0.0f;
  if ((unsigned)h < (unsigned)H_IMG && (unsigned)w < (unsigned)W_IMG) {
    int xidx = (a * H_IMG + h) * W_IMG + w;
    int midx = (c * H_IMG + h) * W_IMG + w;
    float ar = xr[xidx], ai = xi[xidx];
    float br = mr[midx], bi = mi[midx];
    outr = ar * br - ai * bi;
    outi = ar * bi + ai * br;
  }
  wsR[tid] = outr;
  wsI[tid] = outi;
}

// ---------------------------------------------------------------------------
// K2/K3/K5: 512-point FFT of one line per wave (axis 0 = rows, 1 = cols).
//   n = 32*n1 + n2  ->  X[k1 + 16*k2]
//   step1: T[k1][n2]  = sum_n1 F16[k1][n1] * x[n1][n2]        (WMMA, K=16)
//   twid : T *= W512^{n2*k1}
//   step3: R[k1][k2]  = sum_n2 T[k1][n2]  * F32[n2][k2]       (WMMA, K=32)
//   sign = -1 forward, +1 inverse (unnormalized; scaling folded into K6).
// ---------------------------------------------------------------------------
__global__ __launch_bounds__(128)
void fft512_kernel(float* __restrict__ Re, float* __restrict__ Im, int axis, float sign) {
  __shared__ float lsXr[4][512], lsXi[4][512];
  __shared__ float lsTr[4][512], lsTi[4][512];

  const int lane = threadIdx.x & 31;
  const int wv   = threadIdx.x >> 5;
  const int line = blockIdx.x * 4 + wv;     // grid sized exactly: NPLANE*512/4 blocks
  const int m    = lane & 15;               // M (A/C rows) or N (B cols) index
  const int kh   = lane >> 4;               // lane-half selects K pairs / M+8

  int base, stride;
  if (axis == 0) { base = line << 9;                                  stride = 1;  }
  else           { base = ((line >> 9) << 18) + (line & 511);         stride = GW; }

  float* xr = lsXr[wv]; float* xi = lsXi[wv];
  float* tr = lsTr[wv]; float* ti = lsTi[wv];

  // ---- load line into LDS as a 16x32 row-major matrix (stride 32) ----
  #pragma unroll
  for (int i = 0; i < 16; ++i) {
    int idx = lane + 32 * i;
    int g = base + idx * stride;
    xr[idx] = Re[g];
    xi[idx] = Im[g];
  }

  // ---- DFT-16 A-fragments (constant): F16[m][k] = e^{i*sign*2pi*m*k/16} ----
  // a16in holds the pre-negated imaginary fragment for the -Ai*Bi term
  // (float WMMA has no A/B NEG modifier: NEG[1:0] must be 0).
  v2f a16r[4], a16i[4], a16in[4];
  #pragma unroll
  for (int s = 0; s < 4; ++s) {
    #pragma unroll
    for (int j = 0; j < 2; ++j) {
      int k = 4 * s + j + 2 * kh;           // ISA A-layout: VGPR j, lane-half kh
      float ang = sign * (2.0f * PI_F / 16.0f) * (float)(m * k);
      float cv = __cosf(ang), sv = __sinf(ang);
      a16r[s][j]  = cv;
      a16i[s][j]  = sv;
      a16in[s][j] = -sv;
    }
  }

  // ---- step 1 + twiddle ----
  #pragma unroll
  for (int t = 0; t < 2; ++t) {
    const int co = 16 * t;
    v8f accR = {}; v8f accI = {};
    #pragma unroll
    for (int s = 0; s < 4; ++s) {
      v2f bR, bI;
      #pragma unroll
      for (int j = 0; j < 2; ++j) {
        int row = 4 * s + j + 2 * kh;       // ISA B-layout mirrors A over K
        bR[j] = xr[row * 32 + co + m];
        bI[j] = xi[row * 32 + co + m];
      }
      // complex GEMM: accR += Ar*Br + (-Ai)*Bi ; accI += Ar*Bi + Ai*Br
      accR = __builtin_amdgcn_wmma_f32_16x16x4_f32(false, a16r[s],  false, bR, (short)0, accR, false, false);
      accR = __builtin_amdgcn_wmma_f32_16x16x4_f32(false, a16in[s], false, bI, (short)0, accR, false, false);
      accI = __builtin_amdgcn_wmma_f32_16x16x4_f32(false, a16r[s],  false, bI, (short)0, accI, false, false);
      accI = __builtin_amdgcn_wmma_f32_16x16x4_f32(false, a16i[s],  false, bR, (short)0, accI, false, false);
    }
    #pragma unroll
    for (int r = 0; r < 8; ++r) {           // C/D layout: M = r + 8*kh, N = m
      int k1 = r + 8 * kh;
      int n2 = co + m;
      float ang = sign * (2.0f * PI_F / 512.0f) * (float)(k1 * n2);
      float cw = __cosf(ang), sw = __sinf(ang);
      tr[k1 * 32 + n2] = accR[r] * cw - accI[r] * sw;
      ti[k1 * 32 + n2] = accR[r] * sw + accI[r] * cw;
    }
  }

  // ---- step 3: R = T x F32 (K=32), write permuted output k = k1 + 16*k2 ----
  #pragma unroll
  for (int t2 = 0; t2 < 2; ++t2) {
    const int co = 16 * t2;
    v8f accR = {}; v8f accI = {};
    #pragma unroll
    for (int kk = 0; kk < 32; kk += 4) {
      v2f aR, aI, bR, bI, bIn;
      #pragma unroll
      for (int j = 0; j < 2; ++j) {
        int k = kk + j + 2 * kh;
        aR[j] = tr[m * 32 + k];             // A from T: M = m, K = k
        aI[j] = ti[m * 32 + k];
        float ang = sign * (2.0f * PI_F / 32.0f) * (float)(k * (co + m));
        float cv = __cosf(ang), sv = __sinf(ang);
        bR[j]  = cv;                        // F32[k][co+m] (symmetric DFT-32)
        bI[j]  = sv;
        bIn[j] = -sv;
      }
      accR = __builtin_amdgcn_wmma_f32_16x16x4_f32(false, aR, false, bR,  (short)0, accR, false, false);
      accR = __builtin_amdgcn_wmma_f32_16x16x4_f32(false, aI, false, bIn, (short)0, accR, false, false);
      accI = __builtin_amdgcn_wmma_f32_16x16x4_f32(false, aR, false, bI,  (short)0, accI, false, false);
      accI = __builtin_amdgcn_wmma_f32_16x16x4_f32(false, aI, false, bR,  (short)0, accI, false, false);
    }
    #pragma unroll
    for (int r = 0; r < 8; ++r) {
      int k1 = r + 8 * kh;
      int k2 = co + m;
      int g = base + (k1 + 16 * k2) * stride;
      Re[g] = accR[r];
      Im[g] = accI[r];
    }
  }
}

// ---------------------------------------------------------------------------
// K4: per-frequency 5x5 subspace mixing, in place over the a index.
//   Y[c,a,hw] = sum_b K[a,b,hw] * Fx[c,b,hw].  K cached in registers, reused
//   across the 12 coils; per (c,hw) all 5 inputs are read before 5 writes.
// ---------------------------------------------------------------------------
__global__ void subspace_mix_kernel(const float* __restrict__ kr, const float* __restrict__ ki,
                                    float* __restrict__ wsR, float* __restrict__ wsI) {
  int hw = blockIdx.x * blockDim.x + threadIdx.x;       // 0..PLANE_ELEMS-1
  float Kr[A_SUB][A_SUB], Ki[A_SUB][A_SUB];
  #pragma unroll
  for (int a = 0; a < A_SUB; ++a)
    #pragma unroll
    for (int b = 0; b < A_SUB; ++b) {
      int idx = (a * A_SUB + b) * PLANE_ELEMS + hw;
      Kr[a][b] = kr[idx];
      Ki[a][b] = ki[idx];
    }
  for (int c = 0; c < C_COIL; ++c) {
    float fr[A_SUB], fi[A_SUB];
    #pragma unroll
    for (int b = 0; b < A_SUB; ++b) {
      int idx = (c * A_SUB + b) * PLANE_ELEMS + hw;
      fr[b] = wsR[idx];
      fi[b] = wsI[idx];
    }
    #pragma unroll
    for (int a = 0; a < A_SUB; ++a) {
      float yr = 0.0f, yi = 0.0f;
      #pragma unroll
      for (int b = 0; b < A_SUB; ++b) {
        yr += Kr[a][b] * fr[b] - Ki[a][b] * fi[b];
        yi += Kr[a][b] * fi[b] + Ki[a][b] * fr[b];
      }
      int idx = (c * A_SUB + a) * PLANE_ELEMS + hw;
      wsR[idx] = yr;
      wsI[idx] = yi;
    }
  }
}

// ---------------------------------------------------------------------------
// K6: centered crop + conj(mps) coil combine + all scaling.
//   scale = OS^2 * (1/512)_fwd-ortho * (1/512)_inv-ortho = 4 / 512^2
// ---------------------------------------------------------------------------
__global__ void combine_kernel(const float* __restrict__ mr, const float* __restrict__ mi,
                               const float* __restrict__ wsR, const float* __restrict__ wsI,
                               float2* __restrict__ out) {
  int tid = blockIdx.x * blockDim.x + threadIdx.x;      // over A_SUB*H*W
  int w = tid & (W_IMG - 1);
  int h = (tid >> 8) & (H_IMG - 1);
  int a = tid >> 16;
  const float scale = 4.0f / (float)PLANE_ELEMS;
  float sr = 0.0f, si = 0.0f;
  for (int c = 0; c < C_COIL; ++c) {
    int g = (c * A_SUB + a) * PLANE_ELEMS + ((h + 128) << 9) + (w + 128);
    float yr = wsR[g], yi = wsI[g];
    int midx = (c * H_IMG + h) * W_IMG + w;
    float cr = mr[midx], ci = mi[midx];
    sr += yr * cr + yi * ci;                // y * conj(m)
    si += yi * cr - yr * ci;
  }
  out[tid] = make_float2(sr * scale, si * scale);
}

// ---------------------------------------------------------------------------
extern "C" void kernel_launch(void* const* d_in, const int* in_sizes, int n_in,
                              void* d_out, int out_size, void* d_ws, size_t ws_size,
                              hipStream_t stream) {
  const float* xr = (const float*)d_in[0];
  const float* xi = (const float*)d_in[1];
  const float* mr = (const float*)d_in[2];
  const float* mi = (const float*)d_in[3];
  const float* kr = (const float*)d_in[4];
  const float* ki = (const float*)d_in[5];

  // workspace: real/imag planes, 60 * 512^2 floats each (126 MB total)
  float* wsR = (float*)d_ws;
  float* wsI = wsR + (size_t)NPLANE * PLANE_ELEMS;

  const int totalPix = NPLANE * PLANE_ELEMS;            // 15,728,640
  const int lines    = NPLANE * GH;                     // 30,720 (one per wave)

  expand_pad_kernel  <<<totalPix / 256, 256, 0, stream>>>(xr, xi, mr, mi, wsR, wsI);
  fft512_kernel      <<<lines / 4,     128, 0, stream>>>(wsR, wsI, /*axis=*/0, -1.0f);
  fft512_kernel      <<<lines / 4,     128, 0, stream>>>(wsR, wsI, /*axis=*/1, -1.0f);
  subspace_mix_kernel<<<PLANE_ELEMS / 256, 256, 0, stream>>>(kr, ki, wsR, wsI);
  fft512_kernel      <<<lines / 4,     128, 0, stream>>>(wsR, wsI, /*axis=*/1, +1.0f);
  fft512_kernel      <<<lines / 4,     128, 0, stream>>>(wsR, wsI, /*axis=*/0, +1.0f);
  combine_kernel     <<<(A_SUB * H_IMG * W_IMG) / 256, 256, 0, stream>>>(mr, mi, wsR, wsI, (float2*)d_out);
}